// GRU_NAL_Gat_72619307041564
// MI455X (gfx1250) — compile-verified
//
#include <hip/hip_runtime.h>
#include <math.h>

// ---- problem dims ----
constexpr int kB = 16, kT = 12, kN = 2048, kCIN = 64, kCOUT = 64;
constexpr int kDE = 16, kGH = 32, kH = 4;
constexpr int kC2 = 128;   // CIN + COUT
constexpr int kG2 = 128;   // 2 * COUT

typedef float v2f __attribute__((ext_vector_type(2)));
typedef float v8f __attribute__((ext_vector_type(8)));

// ============================================================
// 1) GAT: s[h,n], d[h,n] from node embeddings
// ============================================================
__global__ __launch_bounds__(256) void k_gat_sd(
    const float* __restrict__ E, const float* __restrict__ gW,
    const float* __restrict__ asrc, const float* __restrict__ adst,
    float* __restrict__ sb, float* __restrict__ db) {
  int idx = blockIdx.x * 256 + threadIdx.x;
  if (idx >= kH * kN) return;
  int n = idx % kN;
  int h = idx / kN;
  float e[kDE];
#pragma unroll
  for (int d = 0; d < kDE; ++d) e[d] = E[n * kDE + d];
  float s = 0.f, dd = 0.f;
  for (int f = 0; f < kGH; ++f) {
    float hv = 0.f;
#pragma unroll
    for (int d = 0; d < kDE; ++d) hv += e[d] * gW[(h * kDE + d) * kGH + f];
    s += hv * asrc[h * kGH + f];
    dd += hv * adst[h * kGH + f];
  }
  sb[h * kN + n] = s;
  db[h * kN + n] = dd;
}

// ============================================================
// 2) A[m,:] = mean_h softmax_n( leaky_relu(s[h,m]+d[h,n]) )
//    one block per output row m
// ============================================================
__global__ __launch_bounds__(256) void k_adj_row(
    const float* __restrict__ sb, const float* __restrict__ db,
    float* __restrict__ A) {
  const int m = blockIdx.x;
  const int tid = threadIdx.x;
  __shared__ float dsh[kN];
  __shared__ float red[256];
  float areg[8];
#pragma unroll
  for (int i = 0; i < 8; ++i) areg[i] = 0.f;

  for (int h = 0; h < kH; ++h) {
    const float s = sb[h * kN + m];
    for (int i = tid; i < kN; i += 256) dsh[i] = db[h * kN + i];
    __syncthreads();

    // pass 1: row max
    float mx = -3.0e38f;
#pragma unroll
    for (int i = 0; i < 8; ++i) {
      float e = s + dsh[i * 256 + tid];
      e = (e > 0.f) ? e : 0.2f * e;
      mx = fmaxf(mx, e);
    }
    red[tid] = mx;
    __syncthreads();
    for (int off = 128; off > 0; off >>= 1) {
      if (tid < off) red[tid] = fmaxf(red[tid], red[tid + off]);
      __syncthreads();
    }
    mx = red[0];
    __syncthreads();

    // pass 2: exp + sum
    float ev[8];
    float lsum = 0.f;
#pragma unroll
    for (int i = 0; i < 8; ++i) {
      float e = s + dsh[i * 256 + tid];
      e = (e > 0.f) ? e : 0.2f * e;
      ev[i] = __expf(e - mx);
      lsum += ev[i];
    }
    red[tid] = lsum;
    __syncthreads();
    for (int off = 128; off > 0; off >>= 1) {
      if (tid < off) red[tid] += red[tid + off];
      __syncthreads();
    }
    const float inv = 1.f / red[0];
    __syncthreads();
#pragma unroll
    for (int i = 0; i < 8; ++i) areg[i] += ev[i] * inv;
    __syncthreads();  // before dsh is overwritten next head
  }
#pragma unroll
  for (int i = 0; i < 8; ++i)
    A[(size_t)m * kN + i * 256 + tid] = areg[i] * (1.0f / kH);
}

// ============================================================
// 3a) Node-adaptive weights in WMMA-fragment-native layout:
//     W'[n][k/2][o][k&1] = sum_d E[n,d] * wp[d,k,o_flat]
//     so a lane's K-pair is one aligned 8-byte load.
// ============================================================
__global__ __launch_bounds__(256) void k_nal_w_pair(
    const float* __restrict__ E, const float* __restrict__ wp,
    float* __restrict__ W, int OW) {
  const int per_node = 2 * kC2 * OW;  // 256 * OW
  size_t idx = (size_t)blockIdx.x * 256 + threadIdx.x;
  size_t total = (size_t)kN * per_node;
  if (idx >= total) return;
  int n = (int)(idx / per_node);
  int r = (int)(idx % per_node);
  int ph = r & 1;
  int o = (r >> 1) % OW;
  int kp = r / (2 * OW);
  int k = 2 * kp + ph;
  float acc = 0.f;
#pragma unroll
  for (int d = 0; d < kDE; ++d)
    acc += E[n * kDE + d] * wp[(size_t)d * per_node + (size_t)k * OW + o];
  W[idx] = acc;
}

// 3b) plain pools (biases)
__global__ __launch_bounds__(256) void k_nal_w(
    const float* __restrict__ E, const float* __restrict__ wp,
    float* __restrict__ W, int per_node) {
  size_t idx = (size_t)blockIdx.x * 256 + threadIdx.x;
  size_t total = (size_t)kN * per_node;
  if (idx >= total) return;
  int n = (int)(idx / per_node);
  int r = (int)(idx % per_node);
  float acc = 0.f;
#pragma unroll
  for (int d = 0; d < kDE; ++d)
    acc += E[n * kDE + d] * wp[(size_t)d * per_node + r];
  W[idx] = acc;
}

// ============================================================
// 4) cat = [x_t, state]  (use_r=0)  or  [x_t, r*state] (use_r=1)
// ============================================================
__global__ __launch_bounds__(256) void k_cat(
    const float* __restrict__ x, const float* __restrict__ state,
    const float* __restrict__ zr, float* __restrict__ cat, int t, int use_r) {
  size_t idx = (size_t)blockIdx.x * 256 + threadIdx.x;
  int c = (int)(idx % kC2);
  size_t bn = idx / kC2;
  int n = (int)(bn % kN);
  int b = (int)(bn / kN);
  float v;
  if (c < kCIN) {
    v = x[(((size_t)b * kT + t) * kN + n) * kCIN + c];
  } else {
    float s = state[((size_t)b * kN + n) * kCOUT + (c - kCIN)];
    if (use_r) {
      float r = zr[((size_t)b * kN + n) * kG2 + kCOUT + (c - kCIN)];
      s *= r;
    }
    v = s;
  }
  cat[idx] = v;
}

// ============================================================
// 5) diffusion: Y[b,m,c] = sum_n A[m,n] * X[b,n,c]
//    fp32 WMMA 16x16x4; LDS tiles with K-pair-contiguous layout
//    so every fragment is one aligned ds_load_b64.
// ============================================================
__global__ __launch_bounds__(256) void k_diffusion(
    const float* __restrict__ A, const float* __restrict__ X,
    float* __restrict__ Y) {
  constexpr int BM = 64, BN = 64, BK = 32, BKP = BK + 2;  // even stride
  const int b = blockIdx.z;
  const int m0 = blockIdx.x * BM;
  const int n0 = blockIdx.y * BN;
  const float* Xb = X + (size_t)b * kN * kC2;
  float* Yb = Y + (size_t)b * kN * kC2;

  __shared__ float sA[BM][BKP];  // [m][k]
  __shared__ float sX[BN][BKP];  // [c][k]  (transposed from memory)

  const int tid = threadIdx.x;
  const int wave = tid >> 5;
  const int lane = tid & 31;
  const int wm = wave & 3;   // M sub-tile (16 rows each)
  const int wn = wave >> 2;  // N sub-strip (32 cols each)
  const int lm = lane & 15;
  const int kh = lane >> 4;  // which K pair

  v8f acc0 = {};
  v8f acc1 = {};

  for (int k0 = 0; k0 < kN; k0 += BK) {
    for (int i = tid; i < BM * BK; i += 256) {
      int r = i >> 5, c = i & 31;
      sA[r][c] = A[(size_t)(m0 + r) * kN + (k0 + c)];
    }
    for (int i = tid; i < BK * BN; i += 256) {
      int r = i >> 6, c = i & 63;
      sX[c][r] = Xb[(size_t)(k0 + r) * kC2 + (n0 + c)];
    }
    __syncthreads();
#pragma unroll
    for (int kk = 0; kk < BK; kk += 4) {
      const v2f a = *(const v2f*)&sA[wm * 16 + lm][kk + kh * 2];
      const v2f b0 = *(const v2f*)&sX[wn * 32 + lm][kk + kh * 2];
      const v2f b1 = *(const v2f*)&sX[wn * 32 + 16 + lm][kk + kh * 2];
      acc0 = __builtin_amdgcn_wmma_f32_16x16x4_f32(false, a, false, b0,
                                                   (short)0, acc0, false, false);
      acc1 = __builtin_amdgcn_wmma_f32_16x16x4_f32(false, a, false, b1,
                                                   (short)0, acc1, false, false);
    }
    __syncthreads();
  }
#pragma unroll
  for (int r = 0; r < 8; ++r) {
    int m = m0 + wm * 16 + r + kh * 8;
    int c = n0 + wn * 32 + lm;
    Yb[(size_t)m * kC2 + c] = acc0[r];
    Yb[(size_t)m * kC2 + c + 16] = acc1[r];
  }
}

// ============================================================
// 6) per-node NAL gate GEMM: zr[b,n,:] = sigmoid(xg @ W_g[n] + b_g[n])
//    xg = [cat | diff] (16 x 256) in LDS, W_g[n] in pair-interleaved
//    layout: one global_load_b64 per WMMA B-fragment.
// ============================================================
__global__ __launch_bounds__(256) void k_nal_gates(
    const float* __restrict__ cat, const float* __restrict__ diff,
    const float* __restrict__ Wg, const float* __restrict__ bg,
    float* __restrict__ zr) {
  const int n = blockIdx.x;
  const int tid = threadIdx.x;
  __shared__ float xg[16][260];
  for (int i = tid; i < 16 * 256; i += 256) {
    int m = i >> 8, k = i & 255;
    xg[m][k] = (k < kC2) ? cat[((size_t)m * kN + n) * kC2 + k]
                         : diff[((size_t)m * kN + n) * kC2 + (k - kC2)];
  }
  __syncthreads();

  const int wave = tid >> 5, lane = tid & 31;
  const int o0 = wave * 16;
  const int lm = lane & 15, kh = lane >> 4;
  const v2f* __restrict__ Wv =
      (const v2f*)(Wg + (size_t)n * (2 * kC2 * kG2));  // [kp][o] pairs
  __builtin_prefetch((const void*)(Wv + o0 + lm), 0, 0);

  v8f acc = {};
  for (int k0 = 0; k0 < 256; k0 += 4) {
    const v2f a = *(const v2f*)&xg[lm][k0 + kh * 2];
    const int kp = (k0 >> 1) + kh;
    const v2f bb = Wv[(size_t)kp * kG2 + o0 + lm];
    acc = __builtin_amdgcn_wmma_f32_16x16x4_f32(false, a, false, bb, (short)0,
                                                acc, false, false);
  }
  const float bias = bg[(size_t)n * kG2 + o0 + lm];
#pragma unroll
  for (int r = 0; r < 8; ++r) {
    int m = r + kh * 8;  // batch row
    float v = acc[r] + bias;
    v = 1.0f / (1.0f + __expf(-v));
    zr[((size_t)m * kN + n) * kG2 + o0 + lm] = v;
  }
}

// ============================================================
// 7) per-node NAL candidate GEMM + fused GRU update
//    hc = tanh(xg2 @ W_c[n] + b_c[n]); new = z*state + (1-z)*hc
// ============================================================
__global__ __launch_bounds__(128) void k_nal_cand(
    const float* __restrict__ cat, const float* __restrict__ diff,
    const float* __restrict__ Wc, const float* __restrict__ bc,
    const float* __restrict__ zr, float* __restrict__ state,
    float* __restrict__ inner, int t) {
  const int n = blockIdx.x;
  const int tid = threadIdx.x;
  __shared__ float xg[16][260];
  for (int i = tid; i < 16 * 256; i += 128) {
    int m = i >> 8, k = i & 255;
    xg[m][k] = (k < kC2) ? cat[((size_t)m * kN + n) * kC2 + k]
                         : diff[((size_t)m * kN + n) * kC2 + (k - kC2)];
  }
  __syncthreads();

  const int wave = tid >> 5, lane = tid & 31;
  const int o0 = wave * 16;  // 4 waves cover 64 outputs
  const int lm = lane & 15, kh = lane >> 4;
  const v2f* __restrict__ Wv =
      (const v2f*)(Wc + (size_t)n * (2 * kC2 * kCOUT));  // [kp][o] pairs
  __builtin_prefetch((const void*)(Wv + o0 + lm), 0, 0);

  v8f acc = {};
  for (int k0 = 0; k0 < 256; k0 += 4) {
    const v2f a = *(const v2f*)&xg[lm][k0 + kh * 2];
    const int kp = (k0 >> 1) + kh;
    const v2f bb = Wv[(size_t)kp * kCOUT + o0 + lm];
    acc = __builtin_amdgcn_wmma_f32_16x16x4_f32(false, a, false, bb, (short)0,
                                                acc, false, false);
  }
  const float bias = bc[(size_t)n * kCOUT + o0 + lm];
#pragma unroll
  for (int r = 0; r < 8; ++r) {
    int m = r + kh * 8;  // batch row
    float hc = tanhf(acc[r] + bias);
    size_t sidx = ((size_t)m * kN + n) * kCOUT + o0 + lm;
    float z = zr[((size_t)m * kN + n) * kG2 + o0 + lm];  // z = first half
    float sold = state[sidx];
    float nv = z * sold + (1.f - z) * hc;
    state[sidx] = nv;
    inner[(((size_t)m * kT + t) * kN + n) * kCOUT + o0 + lm] = nv;
  }
}

// ============================================================
// 8) output: temporal conv(k=3,pad=1) + 1x1 residual + ReLU + LayerNorm
//    4 positions per 256-thread block (o = tid%64)
// ============================================================
__global__ __launch_bounds__(256) void k_output(
    const float* __restrict__ inner, const float* __restrict__ x,
    const float* __restrict__ tcw, const float* __restrict__ tcb,
    const float* __restrict__ rcw, const float* __restrict__ rcb,
    const float* __restrict__ lng, const float* __restrict__ lnb,
    float* __restrict__ y) {
  const int tid = threadIdx.x;
  const int p = tid >> 6;
  const int o = tid & 63;
  const size_t pos = (size_t)blockIdx.x * 4 + p;  // over B*T*N
  const int n = (int)(pos % kN);
  const size_t bt = pos / kN;
  const int t = (int)(bt % kT);
  const int b = (int)(bt / kT);

  __shared__ float sin_[4][3][64];
  __shared__ float sx[4][64];
  __shared__ float sred[256];

  sx[p][o] = x[(((size_t)b * kT + t) * kN + n) * kCIN + o];
#pragma unroll
  for (int k = 0; k < 3; ++k) {
    int tt = t + k - 1;
    sin_[p][k][o] = (tt >= 0 && tt < kT)
                        ? inner[(((size_t)b * kT + tt) * kN + n) * kCOUT + o]
                        : 0.f;
  }
  __syncthreads();

  float acc = tcb[o] + rcb[o];
  for (int c = 0; c < 64; ++c) {
    acc += rcw[o * kCIN + c] * sx[p][c];
    const float* w = &tcw[(o * kCOUT + c) * 3];
    acc += w[0] * sin_[p][0][c] + w[1] * sin_[p][1][c] + w[2] * sin_[p][2][c];
  }
  float v = fmaxf(acc, 0.f);

  // LayerNorm over the 64-channel group
  sred[tid] = v;
  __syncthreads();
  for (int off = 32; off > 0; off >>= 1) {
    if (o < off) sred[tid] += sred[tid + off];
    __syncthreads();
  }
  const float mu = sred[p * 64] * (1.0f / 64.0f);
  __syncthreads();
  const float dv = v - mu;
  sred[tid] = dv * dv;
  __syncthreads();
  for (int off = 32; off > 0; off >>= 1) {
    if (o < off) sred[tid] += sred[tid + off];
    __syncthreads();
  }
  const float var = sred[p * 64] * (1.0f / 64.0f);
  float outv = dv * rsqrtf(var + 1e-5f) * lng[o] + lnb[o];
  y[pos * kCOUT + o] = outv;
}

// ============================================================
// 9) flat copy
// ============================================================
__global__ __launch_bounds__(256) void k_copy(const float* __restrict__ src,
                                              float* __restrict__ dst,
                                              size_t n) {
  size_t i = (size_t)blockIdx.x * 256 + threadIdx.x;
  if (i < n) dst[i] = src[i];
}

// ============================================================
// host-side launcher
// workspace layout (floats), total ~554 MiB:
//   A | s | d | Wg | bg | Wc | bc | state | inner | cat | diff | zr
// ============================================================
extern "C" void kernel_launch(void* const* d_in, const int* in_sizes, int n_in,
                              void* d_out, int out_size, void* d_ws,
                              size_t ws_size, hipStream_t stream) {
  const float* x = (const float*)d_in[0];
  const float* init_state = (const float*)d_in[1];
  const float* E = (const float*)d_in[2];
  const float* gat_W = (const float*)d_in[3];
  const float* a_src = (const float*)d_in[4];
  const float* a_dst = (const float*)d_in[5];
  const float* wp_g = (const float*)d_in[6];
  const float* bp_g = (const float*)d_in[7];
  const float* wp_c = (const float*)d_in[8];
  const float* bp_c = (const float*)d_in[9];
  const float* tc_w = (const float*)d_in[10];
  const float* tc_b = (const float*)d_in[11];
  const float* rc_w = (const float*)d_in[12];
  const float* rc_b = (const float*)d_in[13];
  const float* ln_g = (const float*)d_in[14];
  const float* ln_b = (const float*)d_in[15];
  float* out = (float*)d_out;

  float* wsf = (float*)d_ws;
  float* A_ = wsf;
  float* sb_ = A_ + (size_t)kN * kN;
  float* db_ = sb_ + (size_t)kH * kN;
  float* Wg_ = db_ + (size_t)kH * kN;
  float* bg_ = Wg_ + (size_t)kN * 2 * kC2 * kG2;
  float* Wc_ = bg_ + (size_t)kN * kG2;
  float* bc_ = Wc_ + (size_t)kN * 2 * kC2 * kCOUT;
  float* state_ = bc_ + (size_t)kN * kCOUT;
  float* inner_ = state_ + (size_t)kB * kN * kCOUT;
  float* cat_ = inner_ + (size_t)kB * kT * kN * kCOUT;
  float* diff_ = cat_ + (size_t)kB * kN * kC2;
  float* zr_ = diff_ + (size_t)kB * kN * kC2;

  // --- static precompute (time-invariant) ---
  k_gat_sd<<<(kH * kN + 255) / 256, 256, 0, stream>>>(E, gat_W, a_src, a_dst,
                                                      sb_, db_);
  k_adj_row<<<kN, 256, 0, stream>>>(sb_, db_, A_);
  k_nal_w_pair<<<(int)(((size_t)kN * 2 * kC2 * kG2) / 256), 256, 0, stream>>>(
      E, wp_g, Wg_, kG2);
  k_nal_w_pair<<<(int)(((size_t)kN * 2 * kC2 * kCOUT) / 256), 256, 0, stream>>>(
      E, wp_c, Wc_, kCOUT);
  k_nal_w<<<(int)(((size_t)kN * kG2) / 256), 256, 0, stream>>>(E, bp_g, bg_,
                                                               kG2);
  k_nal_w<<<(int)(((size_t)kN * kCOUT) / 256), 256, 0, stream>>>(E, bp_c, bc_,
                                                                 kCOUT);
  k_copy<<<(int)(((size_t)kB * kN * kCOUT + 255) / 256), 256, 0, stream>>>(
      init_state, state_, (size_t)kB * kN * kCOUT);

  // --- GRU scan over T ---
  const dim3 diff_grid(kN / 64, kC2 / 64, kB);
  const int cat_blocks = (int)(((size_t)kB * kN * kC2) / 256);
  for (int t = 0; t < kT; ++t) {
    k_cat<<<cat_blocks, 256, 0, stream>>>(x, state_, zr_, cat_, t, 0);
    k_diffusion<<<diff_grid, 256, 0, stream>>>(A_, cat_, diff_);
    k_nal_gates<<<kN, 256, 0, stream>>>(cat_, diff_, Wg_, bg_, zr_);
    k_cat<<<cat_blocks, 256, 0, stream>>>(x, state_, zr_, cat_, t, 1);
    k_diffusion<<<diff_grid, 256, 0, stream>>>(A_, cat_, diff_);
    k_nal_cand<<<kN, 128, 0, stream>>>(cat_, diff_, Wc_, bc_, zr_, state_,
                                       inner_, t);
  }

  // --- output stage ---
  k_output<<<(int)(((size_t)kB * kT * kN) / 4), 256, 0, stream>>>(
      inner_, x, tc_w, tc_b, rc_w, rc_b, ln_g, ln_b, out);
  k_copy<<<(int)(((size_t)kB * kN * kCOUT + 255) / 256), 256, 0, stream>>>(
      state_, out + (size_t)kB * kT * kN * kCOUT, (size_t)kB * kN * kCOUT);
}